// Word2MatEncoder_59794534695036
// MI455X (gfx1250) — compile-verified
//
#include <hip/hip_runtime.h>
#include <cstdint>

// CDNA5 / gfx1250: wave32, WMMA f32 16x16x4 (A 16x4 = 2 VGPR, B 4x16 = 2 VGPR, C/D 16x16 = 8 VGPR)
typedef __attribute__((ext_vector_type(2))) float v2f;
typedef __attribute__((ext_vector_type(8))) float v8f;
typedef __attribute__((ext_vector_type(4))) unsigned int v4u;
typedef __attribute__((ext_vector_type(4))) int v4i;
typedef __attribute__((ext_vector_type(8))) int v8i;

#define MAT 28          // matrix dim
#define EMB 784         // 28*28
#define SEQ 64          // words per sentence
#define WAVES 8         // waves per block (256 threads)
#define ZPAD 1000       // index inside a W buffer guaranteed to hold 0.0f
#define WSLICE 3072     // floats of LDS per wave: cur(1024) + Wbuf0(1024) + Wbuf1(1024)

// Issue a TDM DMA: copy one 784-float table row (3136 B, contiguous) into LDS.
// D# group0: count=1 | lds_addr | global_addr | type=2.
// D# group1: data_size=4B, tensor_dim0 = tile_dim0 = 784, tile_dim1 = 1,
//            tensor_dim0_stride = tensor_dim1_stride = 784. Groups 2/3 unused (<=2D).
__device__ __forceinline__ void tdm_load_row(uint32_t lds_byte_addr, const float* gsrc) {
  const uint64_t ga = (uint64_t)(uintptr_t)gsrc;
  v4u g0 = { 1u,                                    // count = 1, gather off
             lds_byte_addr,                         // lds_addr
             (uint32_t)ga,                          // global_addr[31:0]
             (uint32_t)(ga >> 32) | 0x80000000u };  // global_addr[56:32] | type=2<<30
  v8i g1 = { (int)(2u << 16),          // data_size = 2 (4 bytes); mask/flags/pad = 0
             (int)(784u << 16),        // tensor_dim0[15:0] << 16 (784)
             (int)(1u << 16),          // tensor_dim0 hi = 0 | tensor_dim1[15:0] = 1
             (int)(784u << 16),        // tensor_dim1 hi = 0 | tile_dim0 = 784
             1,                        // tile_dim1 = 1, tile_dim2 = 0
             784,                      // tensor_dim0_stride lo32
             (int)(784u << 16),        // stride0 hi = 0 | tensor_dim1_stride lo16 = 784
             0 };                      // tensor_dim1_stride hi = 0
  v4i g2 = { 0, 0, 0, 0 };
  v4i g3 = { 0, 0, 0, 0 };
#if __clang_major__ >= 23
  v8i g4 = { 0, 0, 0, 0, 0, 0, 0, 0 };
  __builtin_amdgcn_tensor_load_to_lds(g0, g1, g2, g3, g4, 0);
#else
  __builtin_amdgcn_tensor_load_to_lds(g0, g1, g2, g3, 0);
#endif
}

__global__ __launch_bounds__(256) void w2m_chain_kernel(
    const int* __restrict__ sent,     // [B, 64]
    const float* __restrict__ table,  // [30001, 784]
    float* __restrict__ out)          // [B, 784]
{
  // Per-wave private LDS slices (waves never share; no block barriers needed).
  __shared__ float lds[WAVES * WSLICE];  // 96 KB

  const int lane = threadIdx.x & 31;
  const int wave = threadIdx.x >> 5;
  const int b    = blockIdx.x * WAVES + wave;

  float* curS = &lds[wave * WSLICE];   // 32x32 row-major, stride 32
  float* wBuf0 = curS + 1024;          // [0,784): W row-major; [784,1024): zeros
  float* wBuf1 = curS + 2048;

  const int hh  = lane >> 4;   // half-wave: 0 or 1
  const int l15 = lane & 15;

  // Zero the pad tails once (reads of padded K/N are redirected into them).
  for (int t = 784 + lane; t < 1024; t += 32) { wBuf0[t] = 0.0f; wBuf1[t] = 0.0f; }

  // Wave-uniform LDS byte addresses of the two W buffers (TDM operands live in SGPRs).
  const uint32_t wAddr0 = (uint32_t)__builtin_amdgcn_readfirstlane((int)(uint32_t)(uintptr_t)wBuf0);
  const uint32_t wAddr1 = (uint32_t)__builtin_amdgcn_readfirstlane((int)(uint32_t)(uintptr_t)wBuf1);

  // ---- Prefetch words 0 and 1 via TDM (in-order per-wave completion) ----
  {
    const int w0 = __builtin_amdgcn_readfirstlane(sent[b * SEQ + 0]);
    tdm_load_row(wAddr0, table + (size_t)w0 * EMB);
    const int w1 = __builtin_amdgcn_readfirstlane(sent[b * SEQ + 1]);
    tdm_load_row(wAddr1, table + (size_t)w1 * EMB);
  }
  __builtin_amdgcn_s_wait_tensorcnt(1);  // word 0 landed; word 1 may still fly
  __builtin_amdgcn_wave_barrier();
  asm volatile("" ::: "memory");

  // Accumulator: 2x2 tiles of 16x16 f32 in WMMA C/D layout.
  // Tile (i,j), reg r holds element (M = i*16 + r + 8*hh, N = j*16 + l15).
  v8f acc[2][2];
  for (int i = 0; i < 2; ++i)
    for (int j = 0; j < 2; ++j)
      for (int r = 0; r < 8; ++r) {
        const int M = i * 16 + r + 8 * hh;
        const int N = j * 16 + l15;
        const int idx = (M < MAT && N < MAT) ? (M * MAT + N) : ZPAD;
        acc[i][j][r] = wBuf0[idx];
      }
  __builtin_amdgcn_wave_barrier();

  // ---- Sequential chain: cur = cur @ W[s], s = 1..63, double-buffered TDM ----
  for (int s = 1; s < SEQ; ++s) {
    float* wS = (s & 1) ? wBuf1 : wBuf0;

    // Kick off the DMA for step s+1 into the buffer step s-1 just finished with.
    if (s + 1 < SEQ) {
      const int wn = __builtin_amdgcn_readfirstlane(sent[b * SEQ + s + 1]);
      tdm_load_row((s & 1) ? wAddr0 : wAddr1, table + (size_t)wn * EMB);
    }

    // Stage cur into LDS (32x32, stride 32) so it can be re-read in A layout.
    for (int i = 0; i < 2; ++i)
      for (int j = 0; j < 2; ++j)
        for (int r = 0; r < 8; ++r) {
          const int M = i * 16 + r + 8 * hh;
          const int N = j * 16 + l15;
          curS[M * 32 + N] = acc[i][j][r];
        }

    // Buffer for step s is ready when at most the s+1 prefetch is outstanding.
    if (s + 1 < SEQ) __builtin_amdgcn_s_wait_tensorcnt(1);
    else             __builtin_amdgcn_s_wait_tensorcnt(0);
    __builtin_amdgcn_wave_barrier();
    asm volatile("" ::: "memory");

    v8f n00 = {}, n01 = {}, n10 = {}, n11 = {};

    // K = 32 in 8 steps of 4. EXEC is all-ones here (no divergence): WMMA-legal.
    for (int kk = 0; kk < 8; ++kk) {
      const int K = kk * 4 + 2 * hh;  // lanes 0-15 take K, K+1; lanes 16-31 take K+2, K+3

      // A fragments (cur): lane holds row M = i*16 + l15; (K, K+1) contiguous -> ds_load_b64.
      const v2f a0 = *(const v2f*)&curS[(0 * 16 + l15) * 32 + K];
      const v2f a1 = *(const v2f*)&curS[(1 * 16 + l15) * 32 + K];

      // B fragments (W): lane holds col N = j*16 + l15, rows K and K+1; padded -> zero word.
      auto bfrag = [&](int j) -> v2f {
        const int N  = j * 16 + l15;
        const int i0 = (K + 0 < MAT && N < MAT) ? ((K + 0) * MAT + N) : ZPAD;
        const int i1 = (K + 1 < MAT && N < MAT) ? ((K + 1) * MAT + N) : ZPAD;
        v2f v;
        v.x = wS[i0];
        v.y = wS[i1];
        return v;
      };
      const v2f b0 = bfrag(0);
      const v2f b1 = bfrag(1);

      n00 = __builtin_amdgcn_wmma_f32_16x16x4_f32(false, a0, false, b0, (short)0, n00, false, false);
      n01 = __builtin_amdgcn_wmma_f32_16x16x4_f32(false, a0, false, b1, (short)0, n01, false, false);
      n10 = __builtin_amdgcn_wmma_f32_16x16x4_f32(false, a1, false, b0, (short)0, n10, false, false);
      n11 = __builtin_amdgcn_wmma_f32_16x16x4_f32(false, a1, false, b1, (short)0, n11, false, false);
    }

    acc[0][0] = n00; acc[0][1] = n01;
    acc[1][0] = n10; acc[1][1] = n11;
    __builtin_amdgcn_wave_barrier();
  }

  // ---- Write back the 28x28 interior ----
  for (int i = 0; i < 2; ++i)
    for (int j = 0; j < 2; ++j)
      for (int r = 0; r < 8; ++r) {
        const int M = i * 16 + r + 8 * hh;
        const int N = j * 16 + l15;
        if (M < MAT && N < MAT)
          out[(size_t)b * EMB + M * MAT + N] = acc[i][j][r];
      }
}

extern "C" void kernel_launch(void* const* d_in, const int* in_sizes, int n_in,
                              void* d_out, int out_size, void* d_ws, size_t ws_size,
                              hipStream_t stream) {
  const int* sent    = (const int*)d_in[0];    // [B, 64] int32
  const float* table = (const float*)d_in[1];  // [30001, 784] f32
  float* out         = (float*)d_out;          // [B, 784] f32

  const int batch = in_sizes[0] / SEQ;         // 1024
  const int blocks = batch / WAVES;            // 128 blocks of 8 waves
  w2m_chain_kernel<<<blocks, 256, 0, stream>>>(sent, table, out);
}